// TopDownSegHead_58634893525508
// MI455X (gfx1250) — compile-verified
//
#include <hip/hip_runtime.h>
#include <hip/hip_bf16.h>

// ---------------- problem constants ----------------
#define NQ        300
#define NC        80
#define NQC       (NQ * NC)          // 24000 flat scores
#define NCAND     1000               // NMS candidates
#define CAP       2048               // collection capacity (>= NCAND, pow2 for bitonic)
#define MAXSEG    100
#define NPAIRS    4000000
#define FHW       256
#define MAPSZ     (FHW * FHW)        // 65536
#define NMS_THR   0.65f

// ---------------- workspace layout (bytes) ----------------
#define WS_SCORES  0u                // 24000 * 4  : score bits (uint)
#define WS_CSCORE  98304u            // 1000 * 4   : candidate scores
#define WS_CFEAT   102400u           // 1000 * 4   : candidate feat ids
#define WS_CBOX    106496u           // 1000 * 16  : candidate boxes (label-offset)
#define WS_MASK    131072u           // 1000 * 32 * 4 : suppression bitmask
#define WS_SELS    262144u           // 100 * 4    : selected scores
#define WS_SLOT2U  266240u           // 100 * 4    : slot -> unique map id
#define WS_LUT     270336u           // 300 * 4    : query -> unique map id (-1 = unused)
#define WS_GRID    1048576u          // 100 * 65536 * 4 : encoded logit maps (uint, monotone)
// total ~27.3 MB

typedef float v2f __attribute__((ext_vector_type(2)));
typedef float v8f __attribute__((ext_vector_type(8)));

// fast sigmoid: v_exp_f32 + v_rcp_f32 (both ~1 ulp; matches float tolerance)
__device__ __forceinline__ float dev_sigmoid(float x) {
    return __builtin_amdgcn_rcpf(1.0f + __expf(-x));
}
// monotone float<->uint encoding (total order preserving)
__device__ __forceinline__ unsigned enc_f32(float f) {
    unsigned b = __float_as_uint(f);
    return (b & 0x80000000u) ? ~b : (b | 0x80000000u);
}
__device__ __forceinline__ float dec_f32(unsigned e) {
    unsigned b = (e & 0x80000000u) ? (e ^ 0x80000000u) : ~e;
    return __uint_as_float(b);
}

// ========== Kernel 1: scores + top-1000 selection + sort (1 block, 1024 thr) ==========
__global__ __launch_bounds__(1024) void k_select(const float* __restrict__ cls,
                                                 const float* __restrict__ boxes,
                                                 unsigned char* __restrict__ ws) {
    unsigned* sbits   = (unsigned*)(ws + WS_SCORES);
    float*    cscore  = (float*)(ws + WS_CSCORE);
    int*      cfeat   = (int*)(ws + WS_CFEAT);
    float*    cbox    = (float*)(ws + WS_CBOX);

    __shared__ unsigned long long keys[CAP];
    __shared__ unsigned s_cnt;
    __shared__ int s_pos;

    const int tid = threadIdx.x;

    // sigmoid scores (drop last "background" class column)
    for (int i = tid; i < NQC; i += 1024) {
        int q = i / NC, c = i - q * NC;
        float s = dev_sigmoid(cls[q * (NC + 1) + c]);
        sbits[i] = __float_as_uint(s);   // scores in (0,1): positive -> bit-monotone
    }
    __threadfence_block();
    __syncthreads();

    // binary search threshold T = max u with count(bits >= u) >= NCAND
    unsigned lo = 0u, hi = 0xFFFFFFFFu;
    for (int it = 0; it < 32; ++it) {
        if (hi - lo <= 1u) { __syncthreads(); __syncthreads(); __syncthreads(); continue; }
        unsigned mid = lo + ((hi - lo) >> 1);
        if (tid == 0) s_cnt = 0;
        __syncthreads();
        unsigned local = 0;
        for (int i = tid; i < NQC; i += 1024) local += (sbits[i] >= mid) ? 1u : 0u;
        atomicAdd(&s_cnt, local);
        __syncthreads();
        unsigned g = s_cnt;
        if (g >= (unsigned)NCAND) lo = mid; else hi = mid;
        __syncthreads();
    }
    const unsigned T = lo;

    // collect keys: (score_bits << 32) | ~idx  (desc sort => ties: lower idx first)
    if (tid == 0) s_pos = 0;
    __syncthreads();
    for (int i = tid; i < NQC; i += 1024) {
        unsigned b = sbits[i];
        if (b >= T) {
            int p = atomicAdd(&s_pos, 1);
            if (p < CAP) keys[p] = (((unsigned long long)b) << 32) | (unsigned)(~(unsigned)i);
        }
    }
    __syncthreads();
    int npos = s_pos; if (npos > CAP) npos = CAP;
    for (int i = npos + tid; i < CAP; i += 1024) keys[i] = 0ull;

    // bitonic sort descending
    for (int k = 2; k <= CAP; k <<= 1) {
        for (int j = k >> 1; j > 0; j >>= 1) {
            __syncthreads();
            for (int i = tid; i < CAP; i += 1024) {
                int ixj = i ^ j;
                if (ixj > i) {
                    bool desc = ((i & k) == 0);
                    unsigned long long a = keys[i], b = keys[ixj];
                    bool sw = desc ? (a < b) : (a > b);
                    if (sw) { keys[i] = b; keys[ixj] = a; }
                }
            }
        }
    }
    __syncthreads();

    // emit top-1000 candidates with class-aware box offsets
    for (int t = tid; t < NCAND; t += 1024) {
        unsigned long long key = keys[t];
        unsigned b = (unsigned)(key >> 32);
        unsigned i = (~(unsigned)key) & 0xFFFFFFu;
        int feat = (int)(i / NC);
        int lab  = (int)(i - (unsigned)feat * NC);
        if (feat >= NQ) { feat = 0; lab = 0; b = 0; }   // only for pathological padding
        cscore[t] = __uint_as_float(b);
        cfeat[t]  = feat;
        float off = 4.0f * (float)lab;
        cbox[t * 4 + 0] = boxes[feat * 4 + 0] + off;
        cbox[t * 4 + 1] = boxes[feat * 4 + 1] + off;
        cbox[t * 4 + 2] = boxes[feat * 4 + 2] + off;
        cbox[t * 4 + 3] = boxes[feat * 4 + 3] + off;
    }
}

// ========== Kernel 2: IoU suppression bitmask (1000 blocks x 1024) ==========
__global__ __launch_bounds__(1024) void k_iou(unsigned char* __restrict__ ws) {
    const float* cbox = (const float*)(ws + WS_CBOX);
    unsigned*    mask = (unsigned*)(ws + WS_MASK);
    const int i = blockIdx.x;
    const int j = threadIdx.x;
    const float ix1 = cbox[i * 4 + 0], iy1 = cbox[i * 4 + 1];
    const float ix2 = cbox[i * 4 + 2], iy2 = cbox[i * 4 + 3];
    bool pred = false;
    if (j < NCAND && j > i) {
        float jx1 = cbox[j * 4 + 0], jy1 = cbox[j * 4 + 1];
        float jx2 = cbox[j * 4 + 2], jy2 = cbox[j * 4 + 3];
        float xx1 = fmaxf(ix1, jx1), yy1 = fmaxf(iy1, jy1);
        float xx2 = fminf(ix2, jx2), yy2 = fminf(iy2, jy2);
        float inter = fmaxf(xx2 - xx1, 0.0f) * fmaxf(yy2 - yy1, 0.0f);
        float ai = (ix2 - ix1) * (iy2 - iy1);
        float aj = (jx2 - jx1) * (jy2 - jy1);
        float iou = inter / fmaxf(ai + aj - inter, 1e-9f);
        pred = iou > NMS_THR;
    }
    unsigned bal = __builtin_amdgcn_ballot_w32(pred);   // wave32
    if ((threadIdx.x & 31) == 0) mask[i * 32 + (threadIdx.x >> 5)] = bal;
}

// ========== Kernel 3: greedy NMS + top-100 + unique-map LUT (1 wave) ==========
__global__ __launch_bounds__(32) void k_nms(unsigned char* __restrict__ ws) {
    const float*    cscore = (const float*)(ws + WS_CSCORE);
    const int*      cfeat  = (const int*)(ws + WS_CFEAT);
    const unsigned* mask   = (const unsigned*)(ws + WS_MASK);
    float* sscore = (float*)(ws + WS_SELS);
    int*   slot2u = (int*)(ws + WS_SLOT2U);
    int*   lut    = (int*)(ws + WS_LUT);

    const int lane = threadIdx.x;
    for (int q = lane; q < NQ; q += 32) lut[q] = -1;

    __shared__ int s_feats[MAXSEG];
    __shared__ int s_n;
    if (lane == 0) s_n = 0;
    __syncthreads();

    unsigned rm = 0u;   // this lane's removal word (candidates lane*32 .. lane*32+31)
    for (int i = 0; i < NCAND; ++i) {
        int w = i >> 5, b = i & 31;
        unsigned cur = __shfl(rm, w, 32);
        if (!((cur >> b) & 1u)) {               // candidate i survives
            if (lane == 0 && s_n < MAXSEG) {
                s_feats[s_n] = cfeat[i];
                sscore[s_n]  = cscore[i];
                s_n = s_n + 1;
            }
            rm |= mask[i * 32 + lane];
        }
    }
    __syncthreads();
    if (lane == 0) {
        int n = s_n, uid = 0;
        for (int s = 0; s < MAXSEG; ++s) {
            if (s < n) {
                int f = s_feats[s];
                if (lut[f] < 0) lut[f] = uid++;
                slot2u[s] = lut[f];
            } else {
                sscore[s] = 0.0f;               // matches reference (-inf -> 0 score)
                slot2u[s] = 0;
            }
        }
    }
}

// ========== Kernel 4: init encoded logit grid to enc(-1e4) (uint4 fill) ==========
__global__ __launch_bounds__(256) void k_init(unsigned char* __restrict__ ws) {
    uint4* g = (uint4*)(ws + WS_GRID);
    int i = blockIdx.x * 256 + threadIdx.x;
    unsigned e = enc_f32(-1e4f);
    if (i < (MAXSEG * MAPSZ) / 4) g[i] = make_uint4(e, e, e, e);
}

// ========== Kernel 5: scatter-max of 4M pairs (selected maps only) ==========
__global__ __launch_bounds__(256) void k_scatter(const float* __restrict__ seg,
                                                 const float2* __restrict__ kxy,
                                                 const int* __restrict__ qid,
                                                 const int* __restrict__ refined,
                                                 unsigned char* __restrict__ ws) {
    const int* lut = (const int*)(ws + WS_LUT);
    unsigned*  g   = (unsigned*)(ws + WS_GRID);
    int p = blockIdx.x * 256 + threadIdx.x;
    if (p >= NPAIRS) return;
    int u = lut[qid[p]];
    if (u < 0) return;                          // query not selected: map never read
    float2 xy = kxy[p];
    int px = (int)(xy.x * (float)FHW); px = px < 0 ? 0 : (px > FHW - 1 ? FHW - 1 : px);
    int py = (int)(xy.y * (float)FHW); py = py < 0 ? 0 : (py > FHW - 1 ? FHW - 1 : py);
    float w = seg[p];
    if (refined[p]) w *= 2.0f;                  // REFINED_WEIGHT
    atomicMax(&g[u * MAPSZ + py * FHW + px], enc_f32(w));
}

// ========== Kernel 6: out = diag(score) x sigmoid(maps) via V_WMMA_F32_16X16X4_F32 ==========
// One wave handles a 16(segment) x 16(pixel) tile. D[m,n] = score[m]*P[m,n] is computed
// exactly as four chained 16x16x4 f32 WMMAs with A = 16x4 slices of diag(score).
__global__ __launch_bounds__(256) void k_out(const unsigned char* __restrict__ ws,
                                             float* __restrict__ out) {
    const float*    sscore = (const float*)(ws + WS_SELS);
    const int*      slot2u = (const int*)(ws + WS_SLOT2U);
    const unsigned* g      = (const unsigned*)(ws + WS_GRID);

    const int wid  = blockIdx.x * 8 + (threadIdx.x >> 5);   // 28672 waves total
    const int lane = threadIdx.x & 31;
    const int segTile = wid >> 12;        // / 4096 pixel tiles
    const int pixTile = wid & 4095;
    const int s0   = segTile * 16;
    const int pix0 = pixTile * 16;
    const int hf   = lane >> 4;           // 0: K/M lanes 0-15, 1: lanes 16-31
    const int n    = lane & 15;           // column (pixel) / row (for A)

    const int sA = s0 + n;
    const float sc = (sA < MAXSEG) ? sscore[sA] : 0.0f;     // zero rows beyond 100

    v8f c = {};
#pragma unroll
    for (int j = 0; j < 4; ++j) {
        const int r0 = 4 * j + 2 * hf;    // chunk-local K rows handled by this lane half
        v2f a, b;
        a.x = (n == r0)     ? sc : 0.0f;  // diag(score) slice, columns [4j,4j+4)
        a.y = (n == r0 + 1) ? sc : 0.0f;
        // B rows = P[s0+r0, :], P[s0+r0+1, :] at columns pix0+n (branchless clamped loads)
        int sr0 = s0 + r0;     sr0 = (sr0 < MAXSEG) ? sr0 : 0;
        int sr1 = s0 + r0 + 1; sr1 = (sr1 < MAXSEG) ? sr1 : 0;
        int u0 = slot2u[sr0], u1 = slot2u[sr1];
        b.x = dev_sigmoid(dec_f32(g[u0 * MAPSZ + pix0 + n]));
        b.y = dev_sigmoid(dec_f32(g[u1 * MAPSZ + pix0 + n]));
        c = __builtin_amdgcn_wmma_f32_16x16x4_f32(false, a, false, b,
                                                  (short)0, c, false, false);
    }
#pragma unroll
    for (int r = 0; r < 8; ++r) {         // D layout: VGPR r -> rows r (lanes 0-15), r+8 (16-31)
        int s = s0 + r + 8 * hf;
        if (s < MAXSEG) out[s * MAPSZ + pix0 + n] = c[r];
    }
}

// ---------------- launcher ----------------
extern "C" void kernel_launch(void* const* d_in, const int* in_sizes, int n_in,
                              void* d_out, int out_size, void* d_ws, size_t ws_size,
                              hipStream_t stream) {
    const float*  cls_logits = (const float*)d_in[0];  // [300,81]
    const float*  pred_boxes = (const float*)d_in[1];  // [300,4]
    const float*  seg_logits = (const float*)d_in[2];  // [4M]
    const float2* key_xy     = (const float2*)d_in[3]; // [4M,2]
    const int*    qry_ids    = (const int*)d_in[4];    // [4M]
    const int*    refined    = (const int*)d_in[5];    // [4M] bool-as-int
    // d_in[6] = map_ids: unused by the reference
    unsigned char* ws = (unsigned char*)d_ws;
    float* out = (float*)d_out;                        // [100,256,256]

    k_select <<<1, 1024, 0, stream>>>(cls_logits, pred_boxes, ws);
    k_iou    <<<NCAND, 1024, 0, stream>>>(ws);
    k_nms    <<<1, 32, 0, stream>>>(ws);
    k_init   <<<(MAXSEG * MAPSZ) / 4 / 256, 256, 0, stream>>>(ws);
    k_scatter<<<NPAIRS / 256, 256, 0, stream>>>(seg_logits, key_xy, qry_ids, refined, ws);
    k_out    <<<(7 * 4096) / 8, 256, 0, stream>>>(ws, out);
}